// GlobalMapEncoder_24962349924576
// MI455X (gfx1250) — compile-verified
//
#include <hip/hip_runtime.h>
#include <hip/hip_bf16.h>

#define D_DIM 768
#define B_DIM 64
#define N_DIM 36
#define NSEG  (B_DIM * N_DIM)   // 2304 rows
#define TILES_N (D_DIM / 16)    // 48
#define TILES_M (NSEG / 16)     // 144
#define NTILES  (TILES_M * TILES_N) // 6912

typedef float v2f __attribute__((ext_vector_type(2)));
typedef float v8f __attribute__((ext_vector_type(8)));

// ---------------------------------------------------------------------------
// Kernel 1: pos = gmap_pos_fts[2304,7] @ W_pos[7,768] + b_pos  -> d_out
// One wave (32 lanes) per 16x16 tile; K=7 padded to 8 -> two K=4 WMMA steps.
// V_WMMA_F32_16X16X4_F32 operand layouts (ISA 7.12.2):
//   A 16x4: VGPR0 = K=0 (lanes 0-15) / K=2 (lanes 16-31); VGPR1 = K=1 / K=3
//   B 4x16: same striping (rows split across lane halves)
//   C/D 16x16: VGPR r = row r (lanes 0-15) / row r+8 (lanes 16-31), N = lane%16
// ---------------------------------------------------------------------------
__global__ void pos_linear_wmma(const float* __restrict__ X,    // [2304,7]
                                const float* __restrict__ W,    // [7,768]
                                const float* __restrict__ bias, // [768]
                                float* __restrict__ out)        // [2304,768]
{
    const int wave = threadIdx.x >> 5;
    const int lane = threadIdx.x & 31;
    const int tile = blockIdx.x * 4 + wave;          // < 6912 exactly
    const int tm = tile / TILES_N;
    const int tn = tile % TILES_N;
    const int m0 = tm * 16;
    const int n0 = tn * 16;
    const int l16 = lane & 15;
    const int hi  = (lane >> 4) & 1;                 // lane half
    const int arow = m0 + l16;                       // A: M = lane%16 (both halves)
    const int bcol = n0 + l16;                       // B/C/D: N = lane%16
    const int kb   = hi * 2;                         // K base per lane half

    v8f c = {};
#pragma unroll
    for (int ks = 0; ks < 8; ks += 4) {
        const int k0 = ks + kb;
        const int k1 = k0 + 1;
        v2f a, b;
        a.x = (k0 < 7) ? X[arow * 7 + k0]        : 0.0f;
        a.y = (k1 < 7) ? X[arow * 7 + k1]        : 0.0f;
        b.x = (k0 < 7) ? W[k0 * D_DIM + bcol]    : 0.0f;
        b.y = (k1 < 7) ? W[k1 * D_DIM + bcol]    : 0.0f;
        // 8 args: (neg_a, A, neg_b, B, c_mod, C, reuse_a, reuse_b)
        c = __builtin_amdgcn_wmma_f32_16x16x4_f32(false, a, false, b,
                                                  (short)0, c, false, false);
    }

    const float bv = bias[bcol];
#pragma unroll
    for (int r = 0; r < 8; ++r) {
        const int row = m0 + r + hi * 8;
        out[row * D_DIM + bcol] = c[r] + bv;
    }
}

// ---------------------------------------------------------------------------
// Kernel 2: per segment row (2304 blocks x 256 threads):
//   p = d_out[row]                       (pos from kernel 1)
//   out = LN(p)*g+b + step_table[id] + sum_{e in row} tokens[src[e]] * w[e]
// seg_idx is sorted (b outer, node inner) -> binary search gives edge range,
// so the segment-sum needs no atomics.
// ---------------------------------------------------------------------------
__global__ void fuse_ln_step_segsum(float* __restrict__ out,            // [2304,768]
                                    const float* __restrict__ tokens,   // [B*T*L,768]
                                    const float* __restrict__ ln_g,
                                    const float* __restrict__ ln_b,
                                    const float* __restrict__ step_tab, // [100,768]
                                    const int*   __restrict__ step_ids, // [2304]
                                    const float* __restrict__ wts,
                                    const int*   __restrict__ src_idx,
                                    const int*   __restrict__ seg_idx,
                                    int E)
{
    const int row = blockIdx.x;
    const int tid = threadIdx.x;          // 256 threads, 8 waves
    const int lane = tid & 31;
    const int wv   = tid >> 5;

    float p[3];
#pragma unroll
    for (int i = 0; i < 3; ++i) p[i] = out[row * D_DIM + tid + i * 256];

    // --- LayerNorm reduction (sum, sumsq) over 768 ---
    float s  = p[0] + p[1] + p[2];
    float s2 = p[0] * p[0] + p[1] * p[1] + p[2] * p[2];
#pragma unroll
    for (int off = 16; off > 0; off >>= 1) {
        s  += __shfl_down(s,  off, 32);
        s2 += __shfl_down(s2, off, 32);
    }
    __shared__ float red_s[8], red_s2[8];
    __shared__ int   s_lo, s_hi;
    if (lane == 0) { red_s[wv] = s; red_s2[wv] = s2; }

    // --- thread 0: binary search edge range for this segment ---
    if (tid == 0) {
        int lo = 0, h = E;
        while (lo < h) { int mid = (lo + h) >> 1;
                         if (seg_idx[mid] <  row) lo = mid + 1; else h = mid; }
        s_lo = lo;
        int lo2 = lo; h = E;
        while (lo2 < h) { int mid = (lo2 + h) >> 1;
                          if (seg_idx[mid] <= row) lo2 = mid + 1; else h = mid; }
        s_hi = lo2;
    }
    __syncthreads();
    if (wv == 0) {
        float a = (lane < 8) ? red_s[lane]  : 0.0f;
        float b = (lane < 8) ? red_s2[lane] : 0.0f;
#pragma unroll
        for (int off = 4; off > 0; off >>= 1) {
            a += __shfl_down(a, off, 32);
            b += __shfl_down(b, off, 32);
        }
        if (lane == 0) { red_s[0] = a; red_s2[0] = b; }
    }
    __syncthreads();

    const float inv  = 1.0f / (float)D_DIM;
    const float mu   = red_s[0] * inv;
    const float var  = red_s2[0] * inv - mu * mu;
    const float rstd = rsqrtf(var + 1e-12f);
    const int lo = s_lo, hiE = s_hi;
    const int sid = step_ids[row];

    float acc[3];
#pragma unroll
    for (int i = 0; i < 3; ++i) {
        const int col = tid + i * 256;
        acc[i] = (p[i] - mu) * rstd * ln_g[col] + ln_b[col]
               + step_tab[sid * D_DIM + col];
    }

    // --- segment sum: column-parallel edge accumulation (coalesced rows) ---
    for (int e = lo; e < hiE; ++e) {
        const float w = wts[e];
        const float* trow = tokens + (long)src_idx[e] * D_DIM;
        if (e + 1 < hiE)   // hide gather latency of next token row
            __builtin_prefetch(tokens + (long)src_idx[e + 1] * D_DIM + tid, 0, 0);
#pragma unroll
        for (int i = 0; i < 3; ++i)
            acc[i] = fmaf(trow[tid + i * 256], w, acc[i]);
    }

#pragma unroll
    for (int i = 0; i < 3; ++i)
        out[row * D_DIM + tid + i * 256] = acc[i];
}

// ---------------------------------------------------------------------------
extern "C" void kernel_launch(void* const* d_in, const int* in_sizes, int n_in,
                              void* d_out, int out_size, void* d_ws, size_t ws_size,
                              hipStream_t stream)
{
    const float* traj_embeds = (const float*)d_in[0];   // [64,15,40,768]
    const float* gmap_pos    = (const float*)d_in[1];   // [64,36,7]
    const float* W_pos       = (const float*)d_in[2];   // [7,768]
    const float* b_pos       = (const float*)d_in[3];   // [768]
    const float* ln_g        = (const float*)d_in[4];   // [768]
    const float* ln_b        = (const float*)d_in[5];   // [768]
    const float* step_table  = (const float*)d_in[6];   // [100,768]
    const float* wts         = (const float*)d_in[7];   // [E]
    const int*   src_idx     = (const int*)  d_in[8];   // [E]
    const int*   seg_idx     = (const int*)  d_in[9];   // [E]
    const int*   step_ids    = (const int*)  d_in[10];  // [2304]
    const int E = in_sizes[7];

    float* out = (float*)d_out;                         // [2304,768]

    // Kernel 1: 6912 tiles, 4 waves (128 threads) per block
    pos_linear_wmma<<<NTILES / 4, 128, 0, stream>>>(gmap_pos, W_pos, b_pos, out);

    // Kernel 2: one block per segment row
    fuse_ln_step_segsum<<<NSEG, 256, 0, stream>>>(out, traj_embeds, ln_g, ln_b,
                                                  step_table, step_ids,
                                                  wts, src_idx, seg_idx, E);
}